// SimpleGCN_47373489274966
// MI455X (gfx1250) — compile-verified
//
#include <hip/hip_runtime.h>

// ---------------------------------------------------------------------------
// SimpleGCN for MI455X (gfx1250): fp32 WMMA (exact) for dense transforms,
// atomic scatter for the memory-bound message passing.
// ---------------------------------------------------------------------------

typedef __attribute__((ext_vector_type(2))) float v2f;
typedef __attribute__((ext_vector_type(8))) float v8f;

#define NNODES     100000
#define NEDGES     3200000
#define IN_DIM     128
#define HIDDEN     256
#define NCLASS     10
#define NCLASS_PAD 16
#define NGRAPH     64

// ---------------- degree / normalization ----------------------------------

__global__ void deg_init_kernel(float* deg, int n) {
    int i = blockIdx.x * blockDim.x + threadIdx.x;
    if (i < n) deg[i] = 1.0f;                       // self loop contributes 1
}

__global__ void deg_edge_kernel(const int* __restrict__ dst, float* deg, int e) {
    int i = blockIdx.x * blockDim.x + threadIdx.x;
    if (i < e) atomicAdd(&deg[dst[i]], 1.0f);
}

__global__ void dis_kernel(float* deg, int n) {
    int i = blockIdx.x * blockDim.x + threadIdx.x;
    if (i < n) deg[i] = rsqrtf(deg[i]);             // deg >= 1 always (self loop)
}

// ---------------- weight / bias padding (one small kernel each) ------------

__global__ void pad_w4_kernel(const float* __restrict__ W4, float* __restrict__ W4p) {
    int i = blockIdx.x * blockDim.x + threadIdx.x;  // over HIDDEN*NCLASS_PAD
    if (i >= HIDDEN * NCLASS_PAD) return;
    int k = i / NCLASS_PAD, c = i % NCLASS_PAD;
    W4p[i] = (c < NCLASS) ? W4[k * NCLASS + c] : 0.0f;
}

__global__ void pad_b4_kernel(const float* __restrict__ b4, float* __restrict__ b4p) {
    int i = threadIdx.x;
    if (i < NCLASS_PAD) b4p[i] = (i < NCLASS) ? b4[i] : 0.0f;
}

// ---------------- dense transform: C = A(MxK) * W(KxC), fp32 WMMA ----------
// blockDim = (32, NW). Each wave owns one 16x16 output tile; all waves in a
// block share one staged A tile (same 16 rows), each stages its own W tile.
// K stepped by 16 -> 4 chained V_WMMA_F32_16X16X4_F32 per K-block.
// A frag (16x4): lane L -> M = L&15 ; VGPR0/1 = K {0,1} (lo half) {2,3} (hi).
// B frag (4x16): lane L -> N = L&15 ; same K split by half.
// C/D (16x16):  lane L -> N = L&15 ; VGPR v -> M = v + 8*(L>=16).
// C must be a multiple of 16 (weights pre-padded). No divergence near WMMA.

__global__ void __launch_bounds__(128)
gemm16_wmma_kernel(const float* __restrict__ A, const float* __restrict__ W,
                   float* __restrict__ Cout, int M, int K, int C)
{
    __shared__ float lA[16][20];        // [m][k], row stride 20 -> conflict-free
    __shared__ float lB[4][16][20];     // [wave][k][n]

    const int lane = threadIdx.x;
    const int wid  = threadIdx.y;               // wave id in block
    const int half = lane >> 4;                 // 0: lanes 0-15, 1: lanes 16-31
    const int l15  = lane & 15;
    const int m0   = blockIdx.x * 16;
    const int n0   = (blockIdx.y * blockDim.y + wid) * 16;

    // staging coords: each lane moves two float4s per tile (b128 traffic)
    const int sr = lane >> 2;                   // 0..7
    const int sc = (lane & 3) << 2;             // 0,4,8,12

    v8f acc = {};

    for (int k0 = 0; k0 < K; k0 += 16) {
        if (wid == 0) {
            *(float4*)&lA[sr][sc]     = *(const float4*)(A + (size_t)(m0 + sr) * K + k0 + sc);
            *(float4*)&lA[sr + 8][sc] = *(const float4*)(A + (size_t)(m0 + sr + 8) * K + k0 + sc);
        }
        *(float4*)&lB[wid][sr][sc]     = *(const float4*)(W + (size_t)(k0 + sr) * C + n0 + sc);
        *(float4*)&lB[wid][sr + 8][sc] = *(const float4*)(W + (size_t)(k0 + sr + 8) * C + n0 + sc);
        __syncthreads();

        #pragma unroll
        for (int kk = 0; kk < 16; kk += 4) {
            v2f a, b;
            a.x = lA[l15][kk + 2 * half + 0];
            a.y = lA[l15][kk + 2 * half + 1];
            b.x = lB[wid][kk + 2 * half + 0][l15];
            b.y = lB[wid][kk + 2 * half + 1][l15];
            acc = __builtin_amdgcn_wmma_f32_16x16x4_f32(
                /*neg_a=*/false, a, /*neg_b=*/false, b,
                /*c_mod=*/(short)0, acc, /*reuse_a=*/false, /*reuse_b=*/false);
        }
        __syncthreads();
    }

    #pragma unroll
    for (int v = 0; v < 8; ++v)
        Cout[(size_t)(m0 + v + 8 * half) * C + (n0 + l15)] = acc[v];
}

// ---------------- aggregation ----------------------------------------------

// out[i][c] = bias[c] + dis[i]^2 * ht[i][c]   (self-loop term + bias)
__global__ void selfloop_bias_kernel(const float* __restrict__ ht,
                                     const float* __restrict__ dis,
                                     const float* __restrict__ bias,
                                     float* __restrict__ out,
                                     int n, int C)
{
    int idx = blockIdx.x * blockDim.x + threadIdx.x;
    int total = n * C;
    if (idx >= total) return;
    int i = idx / C, c = idx % C;
    float d = dis[i];
    out[idx] = bias[c] + d * d * ht[idx];
}

// scatter: out[dst] += dis[src]*dis[dst] * ht[src]; C/4 threads per edge
__global__ void edge_scatter_kernel(const int* __restrict__ src,
                                    const int* __restrict__ dst,
                                    const float* __restrict__ dis,
                                    const float* __restrict__ ht,
                                    float* __restrict__ out,
                                    int E, int C)
{
    int tpe = C >> 2;
    long long tid = (long long)blockIdx.x * blockDim.x + threadIdx.x;
    long long total = (long long)E * tpe;
    if (tid >= total) return;
    int e = (int)(tid / tpe);
    int c = ((int)(tid % tpe)) << 2;
    int s = src[e], d = dst[e];
    float w = dis[s] * dis[d];
    float4 h = *(const float4*)(ht + (size_t)s * C + c);
    float* o = out + (size_t)d * C + c;
    atomicAdd(o + 0, w * h.x);
    atomicAdd(o + 1, w * h.y);
    atomicAdd(o + 2, w * h.z);
    atomicAdd(o + 3, w * h.w);
}

__global__ void relu_kernel(float* h, int total) {
    int i = blockIdx.x * blockDim.x + threadIdx.x;
    if (i < total) h[i] = fmaxf(h[i], 0.0f);
}

// ---------------- global mean pool -----------------------------------------

__global__ void pool_zero_kernel(float* sums, float* cnt) {
    int i = blockIdx.x * blockDim.x + threadIdx.x;
    if (i < NGRAPH * NCLASS) sums[i] = 0.0f;
    if (i < NGRAPH) cnt[i] = 0.0f;
}

__global__ void pool_scatter_kernel(const int* __restrict__ batch,
                                    const float* __restrict__ h,
                                    float* sums, float* cnt, int n)
{
    int i = blockIdx.x * blockDim.x + threadIdx.x;
    if (i >= n) return;
    int g = batch[i];
    atomicAdd(&cnt[g], 1.0f);
    #pragma unroll
    for (int c = 0; c < NCLASS; ++c)
        atomicAdd(&sums[g * NCLASS + c], h[(size_t)i * NCLASS_PAD + c]);
}

__global__ void pool_final_kernel(const float* __restrict__ sums,
                                  const float* __restrict__ cnt,
                                  float* __restrict__ out)
{
    int i = blockIdx.x * blockDim.x + threadIdx.x;
    if (i >= NGRAPH * NCLASS) return;
    int g = i / NCLASS;
    out[i] = sums[i] / fmaxf(cnt[g], 1.0f);
}

// ---------------------------------------------------------------------------

extern "C" void kernel_launch(void* const* d_in, const int* in_sizes, int n_in,
                              void* d_out, int out_size, void* d_ws, size_t ws_size,
                              hipStream_t stream)
{
    (void)in_sizes; (void)n_in; (void)out_size; (void)ws_size;

    const float* x     = (const float*)d_in[0];
    const int*   ei    = (const int*)  d_in[1];
    const int*   batch = (const int*)  d_in[2];
    const float* W1 = (const float*)d_in[3];  const float* b1 = (const float*)d_in[4];
    const float* W2 = (const float*)d_in[5];  const float* b2 = (const float*)d_in[6];
    const float* W3 = (const float*)d_in[7];  const float* b3 = (const float*)d_in[8];
    const float* W4 = (const float*)d_in[9];  const float* b4 = (const float*)d_in[10];

    const int* src = ei;            // edge_index[0]
    const int* dst = ei + NEDGES;   // edge_index[1]

    // workspace carve-out
    char* ws = (char*)d_ws;
    size_t off = 0;
    auto take = [&](size_t bytes) -> char* {
        char* p = ws + off;
        off += (bytes + 255) & ~(size_t)255;
        return p;
    };
    float* dis  = (float*)take((size_t)NNODES * 4);
    float* bufT = (float*)take((size_t)NNODES * HIDDEN * 4);  // transform output
    float* bufH = (float*)take((size_t)NNODES * HIDDEN * 4);  // aggregated output
    float* W4p  = (float*)take((size_t)HIDDEN * NCLASS_PAD * 4);
    float* b4p  = (float*)take((size_t)NCLASS_PAD * 4);
    float* sums = (float*)take((size_t)NGRAPH * NCLASS * 4);
    float* cnt  = (float*)take((size_t)NGRAPH * 4);

    const int B = 256;
    // --- normalization coefficients + padded tail weights ---
    deg_init_kernel<<<(NNODES + B - 1) / B, B, 0, stream>>>(dis, NNODES);
    deg_edge_kernel<<<(NEDGES + B - 1) / B, B, 0, stream>>>(dst, dis, NEDGES);
    dis_kernel<<<(NNODES + B - 1) / B, B, 0, stream>>>(dis, NNODES);
    pad_w4_kernel<<<(HIDDEN * NCLASS_PAD + B - 1) / B, B, 0, stream>>>(W4, W4p);
    pad_b4_kernel<<<1, NCLASS_PAD, 0, stream>>>(b4, b4p);

    const dim3 gGemm(NNODES / 16, HIDDEN / 64);   // 4 col tiles per block
    const dim3 bGemm(32, 4);
    const int  nNC        = NNODES * HIDDEN;
    const long long eT256 = (long long)NEDGES * (HIDDEN / 4);
    const int  eB256      = (int)((eT256 + B - 1) / B);

    // ---- layer 1: x @ W1 -> aggregate -> relu ----
    gemm16_wmma_kernel<<<gGemm, bGemm, 0, stream>>>(x, W1, bufT, NNODES, IN_DIM, HIDDEN);
    selfloop_bias_kernel<<<(nNC + B - 1) / B, B, 0, stream>>>(bufT, dis, b1, bufH, NNODES, HIDDEN);
    edge_scatter_kernel<<<eB256, B, 0, stream>>>(src, dst, dis, bufT, bufH, NEDGES, HIDDEN);
    relu_kernel<<<(nNC + B - 1) / B, B, 0, stream>>>(bufH, nNC);

    // ---- layer 2 ----
    gemm16_wmma_kernel<<<gGemm, bGemm, 0, stream>>>(bufH, W2, bufT, NNODES, HIDDEN, HIDDEN);
    selfloop_bias_kernel<<<(nNC + B - 1) / B, B, 0, stream>>>(bufT, dis, b2, bufH, NNODES, HIDDEN);
    edge_scatter_kernel<<<eB256, B, 0, stream>>>(src, dst, dis, bufT, bufH, NEDGES, HIDDEN);
    relu_kernel<<<(nNC + B - 1) / B, B, 0, stream>>>(bufH, nNC);

    // ---- layer 3 ----
    gemm16_wmma_kernel<<<gGemm, bGemm, 0, stream>>>(bufH, W3, bufT, NNODES, HIDDEN, HIDDEN);
    selfloop_bias_kernel<<<(nNC + B - 1) / B, B, 0, stream>>>(bufT, dis, b3, bufH, NNODES, HIDDEN);
    edge_scatter_kernel<<<eB256, B, 0, stream>>>(src, dst, dis, bufT, bufH, NEDGES, HIDDEN);
    relu_kernel<<<(nNC + B - 1) / B, B, 0, stream>>>(bufH, nNC);

    // ---- layer 4: HIDDEN -> NCLASS (padded to 16 cols), no relu ----
    const dim3 gGemm4(NNODES / 16, 1);
    const dim3 bGemm4(32, 1);
    gemm16_wmma_kernel<<<gGemm4, bGemm4, 0, stream>>>(bufH, W4p, bufT, NNODES, HIDDEN, NCLASS_PAD);
    const int nN16 = NNODES * NCLASS_PAD;
    selfloop_bias_kernel<<<(nN16 + B - 1) / B, B, 0, stream>>>(bufT, dis, b4p, bufH, NNODES, NCLASS_PAD);
    const long long eT16 = (long long)NEDGES * (NCLASS_PAD / 4);
    edge_scatter_kernel<<<(int)((eT16 + B - 1) / B), B, 0, stream>>>(src, dst, dis, bufT, bufH, NEDGES, NCLASS_PAD);

    // ---- global mean pool ----
    pool_zero_kernel<<<(NGRAPH * NCLASS + B - 1) / B, B, 0, stream>>>(sums, cnt);
    pool_scatter_kernel<<<(NNODES + B - 1) / B, B, 0, stream>>>(batch, bufH, sums, cnt, NNODES);
    pool_final_kernel<<<(NGRAPH * NCLASS + B - 1) / B, B, 0, stream>>>(sums, cnt, (float*)d_out);
}